// SMFNet_37993280701086
// MI455X (gfx1250) — compile-verified
//
#include <hip/hip_runtime.h>

typedef __attribute__((ext_vector_type(2))) float v2f;
typedef __attribute__((ext_vector_type(4))) float v4f;
typedef __attribute__((ext_vector_type(8))) float v8f;

#define TILE_ROWS 16
#define XSTRIDE   132   // padded LDS row stride (floats): 132 % 64 banks == 4 -> conflict-free frags
#define WSTRIDE   132
#define WAVES     4
#define THREADS   128

__global__ __launch_bounds__(THREADS) void smf_wmma_kernel(
    const float* __restrict__ X,
    const float* __restrict__ Wf,
    const float* __restrict__ Wg,
    float* __restrict__ out,
    int N)
{
    __shared__ float xlds[WAVES * TILE_ROWS * XSTRIDE]; // per-wave 16x128 X tiles (padded)
    __shared__ float wlds[5 * WSTRIDE];                 // rows: Wf0, Wf1, Wg0, Wg1, zeros
    __shared__ float res [WAVES * TILE_ROWS * 4];       // relu'd {f0,f1,v0,v1} per row

    const int tid  = threadIdx.x;
    const int lane = tid & 31;
    const int wave = tid >> 5;

    // ---- stage weights (+ zero row for B columns n>=4) into LDS ----
    if (tid < 128) {
        wlds[0 * WSTRIDE + tid] = Wf[tid];         // Wf row 0
        wlds[1 * WSTRIDE + tid] = Wf[128 + tid];   // Wf row 1
        wlds[2 * WSTRIDE + tid] = Wg[tid];         // Wg row 0
        wlds[3 * WSTRIDE + tid] = Wg[128 + tid];   // Wg row 1
        wlds[4 * WSTRIDE + tid] = 0.0f;            // zero row
    }

    const int row0 = (blockIdx.x * WAVES + wave) * TILE_ROWS;

    // ---- stage this wave's 16x128 f32 X tile into LDS (coalesced NT b128) ----
    float* xw = &xlds[wave * TILE_ROWS * XSTRIDE];
    {
        const float* gx = X + (size_t)row0 * 128;
        #pragma unroll
        for (int it = 0; it < 16; ++it) {
            int idx = it * 32 + lane;   // float4 slot within the 16x128 tile
            int r   = idx >> 5;         // 32 float4 per row
            int c4  = idx & 31;
            v4f v = __builtin_nontemporal_load((const v4f*)(gx + (size_t)r * 128 + c4 * 4));
            *(v4f*)(xw + r * XSTRIDE + c4 * 4) = v;
        }
    }
    __syncthreads();

    // ---- D(16x16) += A(16x4 of X) x B(4x16 of W), K = 0..127 step 4 ----
    // A layout: lanes 0-15 hold (M=lane, K, K+1); lanes 16-31 hold (M=lane-16, K+2, K+3)
    // B layout: lanes 0-15 hold (K, N=lane); lanes 16-31 hold (K+2, N=lane-16)
    const int m    = lane & 15;
    const int koff = (lane >> 4) ? 2 : 0;
    const int n    = lane & 15;
    const float* arow = xw + m * XSTRIDE + koff;
    const float* brow = wlds + (n < 4 ? n : 4) * WSTRIDE + koff;

    v8f acc = {};
    #pragma unroll
    for (int kk = 0; kk < 128; kk += 4) {
        v2f a = *(const v2f*)(arow + kk);
        v2f b = *(const v2f*)(brow + kk);
        acc = __builtin_amdgcn_wmma_f32_16x16x4_f32(false, a, false, b,
                                                    (short)0, acc, false, false);
    }

    // ---- tile seam: vnext_k = relu( X[(row0+16)%N] . Wg[k] ), wave-cooperative ----
    const size_t nrow = (size_t)((row0 + TILE_ROWS) % N);   // wrap like jnp.roll
    v4f xq = __builtin_nontemporal_load((const v4f*)(X + nrow * 128 + lane * 4));
    v4f w2 = *(const v4f*)(wlds + 2 * WSTRIDE + lane * 4);
    v4f w3 = *(const v4f*)(wlds + 3 * WSTRIDE + lane * 4);
    float p0 = xq.x * w2.x + xq.y * w2.y + xq.z * w2.z + xq.w * w2.w;
    float p1 = xq.x * w3.x + xq.y * w3.y + xq.z * w3.z + xq.w * w3.w;
    #pragma unroll
    for (int off = 16; off >= 1; off >>= 1) {
        p0 += __shfl_xor(p0, off, 32);
        p1 += __shfl_xor(p1, off, 32);
    }
    const float vnext0 = fmaxf(p0, 0.0f);
    const float vnext1 = fmaxf(p1, 0.0f);

    // ---- stage relu'd WMMA results: res[row][n] for n in {f0,f1,v0,v1} ----
    float* rw = &res[wave * TILE_ROWS * 4];
    if (n < 4) {
        const int mbase = (lane >> 4) ? 8 : 0;  // C/D: lanes 16-31 carry M = r+8
        #pragma unroll
        for (int r = 0; r < 8; ++r) {
            rw[(mbase + r) * 4 + n] = fmaxf(acc[r], 0.0f);
        }
    }
    __syncthreads();

    // ---- combine: out[i][k] = F0[i]*Vk[i] + F1[i]*Vk[i+1] ----
    {
        const int r = lane >> 1;   // row within tile (2 lanes per row)
        const int k = lane & 1;    // output component
        float f0 = rw[r * 4 + 0];
        float f1 = rw[r * 4 + 1];
        float vk = rw[r * 4 + 2 + k];
        float vn = (r < 15) ? rw[(r + 1) * 4 + 2 + k] : (k ? vnext1 : vnext0);
        float o  = f0 * vk + f1 * vn;
        __builtin_nontemporal_store(o, out + (size_t)(row0 + r) * 2 + k);
    }
}

extern "C" void kernel_launch(void* const* d_in, const int* in_sizes, int n_in,
                              void* d_out, int out_size, void* d_ws, size_t ws_size,
                              hipStream_t stream) {
    const float* X  = (const float*)d_in[0];
    const float* Wf = (const float*)d_in[1];
    const float* Wg = (const float*)d_in[2];
    float* out = (float*)d_out;

    const int N      = in_sizes[0] / 128;      // 1,000,000
    const int blocks = N / (WAVES * TILE_ROWS); // 64 rows per block -> 15625 blocks
    smf_wmma_kernel<<<blocks, THREADS, 0, stream>>>(X, Wf, Wg, out, N);
}